// SimilarityGuidedSampling_7911329759394
// MI455X (gfx1250) — compile-verified
//
#include <hip/hip_runtime.h>
#include <math.h>

typedef __attribute__((ext_vector_type(2))) float v2f;
typedef __attribute__((ext_vector_type(4))) float v4f;
typedef __attribute__((ext_vector_type(8))) float v8f;

#define B_    8
#define C_    256
#define T_    32
#define HW_   784
#define HW4_  196        // HW_/4 float4 per t-slice (784*4B = 196*16B, 16B aligned)
#define HID_  512
#define EMB_  32
#define NBIN  4
#define NN    256        // B_*T_ : GEMM N dimension
#define EPSV  1e-12f

// ---------------- Kernel 1: spatial mean pool ----------------
// x: (B,C,T,H,W) -> pooled: (C, B*T)  laid out K x N for GEMM1
// One wave per t-slice: b128 loads + shfl reduction (no LDS, no barriers).
__global__ void pool_kernel(const float* __restrict__ x, float* __restrict__ pooled) {
    int bc = blockIdx.x;                  // b*C + c
    int b = bc / C_, c = bc % C_;
    int wave = threadIdx.x >> 5;
    int lane = threadIdx.x & 31;
    for (int t = wave; t < T_; t += 8) {
        const v4f* p4 = (const v4f*)(x + ((size_t)bc * T_ + t) * HW_);
        float s = 0.f;
        for (int i = lane; i < HW4_; i += 32) {
            v4f v = p4[i];
            s += v.x + v.y + v.z + v.w;
        }
        for (int off = 16; off > 0; off >>= 1) s += __shfl_xor(s, off, 32);
        if (lane == 0)
            pooled[c * NN + b * T_ + t] = s * (1.0f / (float)HW_);
    }
}

// ---------------- Kernel 2: GEMM1 (W1 x pooled) + bias + hswish ----------------
// W1: (512,256) row-major (M x K); pooled: (256, 256) (K x N); H: (512,256) (M x N)
__global__ void gemm1_kernel(const float* __restrict__ W1, const float* __restrict__ b1,
                             const float* __restrict__ P, float* __restrict__ H) {
    int n0 = blockIdx.x * 16;
    int m0 = blockIdx.y * 16;
    int lane = threadIdx.x;
    int half = lane >> 4, lo = lane & 15;
    v8f acc = {};
    for (int k0 = 0; k0 < C_; k0 += 4) {
        int ka = k0 + 2 * half;
        v2f a, b;
        a.x = W1[(m0 + lo) * C_ + ka];
        a.y = W1[(m0 + lo) * C_ + ka + 1];
        b.x = P[ka * NN + n0 + lo];
        b.y = P[(ka + 1) * NN + n0 + lo];
        acc = __builtin_amdgcn_wmma_f32_16x16x4_f32(false, a, false, b,
                                                    (short)0, acc, false, false);
    }
#pragma unroll
    for (int r = 0; r < 8; ++r) {
        int m = m0 + r + 8 * half;
        int n = n0 + lo;
        float v = acc[r] + b1[m];
        float g = fminf(fmaxf(v + 3.0f, 0.0f), 6.0f);
        H[m * NN + n] = v * g * (1.0f / 6.0f);
    }
}

// ---------------- Kernel 3: GEMM2 (W2 x H) + bias ----------------
// W2: (32,512) (M x K); H: (512,256) (K x N); E: (32,256) (M x N)
__global__ void gemm2_kernel(const float* __restrict__ W2, const float* __restrict__ b2,
                             const float* __restrict__ Hm, float* __restrict__ E) {
    int n0 = blockIdx.x * 16;
    int m0 = blockIdx.y * 16;
    int lane = threadIdx.x;
    int half = lane >> 4, lo = lane & 15;
    v8f acc = {};
    for (int k0 = 0; k0 < HID_; k0 += 4) {
        int ka = k0 + 2 * half;
        v2f a, b;
        a.x = W2[(m0 + lo) * HID_ + ka];
        a.y = W2[(m0 + lo) * HID_ + ka + 1];
        b.x = Hm[ka * NN + n0 + lo];
        b.y = Hm[(ka + 1) * NN + n0 + lo];
        acc = __builtin_amdgcn_wmma_f32_16x16x4_f32(false, a, false, b,
                                                    (short)0, acc, false, false);
    }
#pragma unroll
    for (int r = 0; r < 8; ++r) {
        int m = m0 + r + 8 * half;
        int n = n0 + lo;
        E[m * NN + n] = acc[r] + b2[m];
    }
}

// ---------------- Kernel 4: per-batch grouping + softmax weights ----------------
// E: (32, 256); wgt out: (B, T, NBIN)
__global__ void weights_kernel(const float* __restrict__ E, float* __restrict__ wgt) {
    int b = blockIdx.x;
    int t = threadIdx.x;                      // one wave32, lane == t
    __shared__ float sne[EMB_][T_ + 1];       // normalized embds, [e][t]
    __shared__ float ns_s[T_];                // 31 used
    __shared__ int   grp_s[T_];
    __shared__ float cns[EMB_][NBIN];         // normalized centers

    // load + l2-normalize this lane's column of embds
    float ss = 0.f;
    float col[EMB_];
#pragma unroll
    for (int j = 0; j < EMB_; ++j) {
        float v = E[j * NN + b * T_ + t];
        col[j] = v; ss += v * v;
    }
    float inv = 1.0f / fmaxf(sqrtf(ss), EPSV);
#pragma unroll
    for (int j = 0; j < EMB_; ++j) sne[j][t] = col[j] * inv;
    __syncthreads();

    // neighbor similarities ns[0..30]
    if (t >= 1) {
        float d = 0.f;
#pragma unroll
        for (int j = 0; j < EMB_; ++j) d += sne[j][t] * sne[j][t - 1];
        ns_s[t - 1] = d;
    }
    __syncthreads();

    // lane 0: 3rd-smallest threshold, cumsum groups
    if (t == 0) {
        float buf[T_ - 1];
        for (int i = 0; i < T_ - 1; ++i) buf[i] = ns_s[i];
        float th = 0.f;
        for (int pass = 0; pass < NBIN - 1; ++pass) {
            float mn = 3.4e38f; int idx = 0;
            for (int i = 0; i < T_ - 1; ++i)
                if (buf[i] < mn) { mn = buf[i]; idx = i; }
            buf[idx] = 3.4e38f;
            th = mn;
        }
        int g = 0; grp_s[0] = 0;
        for (int i = 1; i < T_; ++i) {
            if (!(ns_s[i - 1] > th)) g++;
            grp_s[i] = g;
        }
    }
    __syncthreads();

    // centers: lane j handles embedding dim j
    int j = t;
#pragma unroll
    for (int k = 0; k < NBIN; ++k) {
        float s = 0.f; int cnt = 0;
        for (int tt = 0; tt < T_; ++tt)
            if (grp_s[tt] == k) { s += sne[j][tt]; cnt++; }
        float avg = s / (float)(cnt > 0 ? cnt : 1);
        float sq = avg * avg;
        for (int off = 16; off > 0; off >>= 1) sq += __shfl_xor(sq, off, 32);
        cns[j][k] = avg / fmaxf(sqrtf(sq), EPSV);
    }
    __syncthreads();

    // similarities + softmax over k for this lane's t
    float w[NBIN], mx = -3.4e38f;
#pragma unroll
    for (int k = 0; k < NBIN; ++k) {
        float s = 0.f;
        for (int jj = 0; jj < EMB_; ++jj) s += sne[jj][t] * cns[jj][k];
        w[k] = 5.0f * s;
        mx = fmaxf(mx, w[k]);
    }
    float den = 0.f;
#pragma unroll
    for (int k = 0; k < NBIN; ++k) { w[k] = __expf(w[k] - mx); den += w[k]; }
#pragma unroll
    for (int k = 0; k < NBIN; ++k) w[k] /= den;

    // per-k sum over t -> scale -> normalized weights
#pragma unroll
    for (int k = 0; k < NBIN; ++k) {
        float s = w[k];
        for (int off = 16; off > 0; off >>= 1) s += __shfl_xor(s, off, 32);
        float scale = (s > 0.f) ? 1.0f / s : 1.0f;
        wgt[b * (T_ * NBIN) + t * NBIN + k] = w[k] * scale;
    }
}

// ---------------- Kernel 5: out[b,c,k,h,w] = sum_t x[b,c,t,h,w] * wgt[b,t,k] ----------------
// 196 threads/block: one float4 pixel-column per thread, b128 NT loads/stores.
__global__ void output_kernel(const float* __restrict__ x, const float* __restrict__ wgt,
                              float* __restrict__ out) {
    int bc = blockIdx.x;
    int b = bc / C_;
    __shared__ float w[T_ * NBIN];
    if (threadIdx.x < T_ * NBIN) w[threadIdx.x] = wgt[b * T_ * NBIN + threadIdx.x];
    __syncthreads();
    const v4f* xp4 = (const v4f*)(x + (size_t)bc * T_ * HW_);
    v4f* op4 = (v4f*)(out + (size_t)bc * NBIN * HW_);
    int p = threadIdx.x;                      // 0..195
    v4f a0 = {}, a1 = {}, a2 = {}, a3 = {};
#pragma unroll
    for (int t = 0; t < T_; ++t) {
        v4f xv = __builtin_nontemporal_load(&xp4[t * HW4_ + p]);   // last use of x
        float w0 = w[t * 4 + 0], w1 = w[t * 4 + 1], w2 = w[t * 4 + 2], w3 = w[t * 4 + 3];
        a0 += xv * w0;
        a1 += xv * w1;
        a2 += xv * w2;
        a3 += xv * w3;
    }
    __builtin_nontemporal_store(a0, &op4[0 * HW4_ + p]);
    __builtin_nontemporal_store(a1, &op4[1 * HW4_ + p]);
    __builtin_nontemporal_store(a2, &op4[2 * HW4_ + p]);
    __builtin_nontemporal_store(a3, &op4[3 * HW4_ + p]);
}

extern "C" void kernel_launch(void* const* d_in, const int* in_sizes, int n_in,
                              void* d_out, int out_size, void* d_ws, size_t ws_size,
                              hipStream_t stream) {
    const float* x  = (const float*)d_in[0];
    const float* W1 = (const float*)d_in[1];
    const float* b1 = (const float*)d_in[2];
    const float* W2 = (const float*)d_in[3];
    const float* b2 = (const float*)d_in[4];
    float* out = (float*)d_out;

    float* ws     = (float*)d_ws;
    float* pooled = ws;                       // C_*NN       = 65536 floats
    float* Hbuf   = pooled + C_ * NN;         // HID_*NN     = 131072 floats
    float* Ebuf   = Hbuf + HID_ * NN;         // EMB_*NN     = 8192 floats
    float* wgt    = Ebuf + EMB_ * NN;         // B_*T_*NBIN  = 1024 floats

    pool_kernel<<<B_ * C_, 256, 0, stream>>>(x, pooled);
    gemm1_kernel<<<dim3(NN / 16, HID_ / 16), 32, 0, stream>>>(W1, b1, pooled, Hbuf);
    gemm2_kernel<<<dim3(NN / 16, EMB_ / 16), 32, 0, stream>>>(W2, b2, Hbuf, Ebuf);
    weights_kernel<<<B_, 32, 0, stream>>>(Ebuf, wgt);
    output_kernel<<<B_ * C_, 196, 0, stream>>>(x, wgt, out);
}